// MolTransformer_4801773437004
// MI455X (gfx1250) — compile-verified
//
#include <hip/hip_runtime.h>
#include <hip/hip_bf16.h>

typedef __attribute__((ext_vector_type(2))) float v2f;
typedef __attribute__((ext_vector_type(8))) float v8f;

// ---------------------------------------------------------------------------
// Generic f32 WMMA GEMM:  C[M x N] = A[M x K] @ (BT ? W[N x K]^T : B[K x N])
// K is compile-time: full k-steps are unguarded vector loads; only the
// K%4 tail uses clamped-address loads + value selects (no EXEC branches).
// Wave = one 16x16 C tile; block = 128 threads = 4 waves covering 64 rows.
// grid = (M/64, N/16, batch).
// ---------------------------------------------------------------------------
template <int K, bool BT, bool BIAS, bool RELU>
__global__ __launch_bounds__(128) void gemm_f32_wmma(
    const float* __restrict__ A, int lda, long sA,
    const float* __restrict__ B, int ldb, long sB,
    float* __restrict__ C, int ldc, long sC,
    const float* __restrict__ bias)
{
    const int lane = threadIdx.x & 31;
    const int wave = threadIdx.x >> 5;
    const long zB = (long)blockIdx.z * sB;
    const long zC = (long)blockIdx.z * sC;

    const int row0 = blockIdx.x * 64 + wave * 16;
    const int col0 = blockIdx.y * 16;
    const int mrow = row0 + (lane & 15);
    const int ncol = col0 + (lane & 15);
    const int kh   = (lane >> 4) << 1;        // 0 or 2

    const float* Ap = A + (long)blockIdx.z * sA + (long)mrow * lda + kh;
    const float* Bp = BT ? (B + zB + (long)ncol * ldb + kh)
                         : (B + zB + ncol);

    constexpr int KFULL = K & ~3;             // complete 4-wide steps
    constexpr int KF8   = KFULL & ~7;         // pairs of steps

    v8f acc0, acc1;
    #pragma unroll
    for (int r = 0; r < 8; ++r) { acc0[r] = 0.0f; acc1[r] = 0.0f; }

    #pragma unroll 4
    for (int k0 = 0; k0 < KF8; k0 += 8) {
        v2f a0 = *(const v2f*)(Ap + k0);
        v2f a1 = *(const v2f*)(Ap + k0 + 4);
        v2f b0, b1;
        if (BT) {
            b0 = *(const v2f*)(Bp + k0);
            b1 = *(const v2f*)(Bp + k0 + 4);
        } else {
            b0.x = Bp[(long)(k0 + kh)     * ldb];
            b0.y = Bp[(long)(k0 + kh + 1) * ldb];
            b1.x = Bp[(long)(k0 + kh + 4) * ldb];
            b1.y = Bp[(long)(k0 + kh + 5) * ldb];
        }
        acc0 = __builtin_amdgcn_wmma_f32_16x16x4_f32(
            false, a0, false, b0, (short)0, acc0, false, false);
        acc1 = __builtin_amdgcn_wmma_f32_16x16x4_f32(
            false, a1, false, b1, (short)0, acc1, false, false);
    }
    if constexpr (KF8 != KFULL) {             // one leftover full step
        v2f a0 = *(const v2f*)(Ap + KF8);
        v2f b0;
        if (BT) b0 = *(const v2f*)(Bp + KF8);
        else {
            b0.x = Bp[(long)(KF8 + kh)     * ldb];
            b0.y = Bp[(long)(KF8 + kh + 1) * ldb];
        }
        acc0 = __builtin_amdgcn_wmma_f32_16x16x4_f32(
            false, a0, false, b0, (short)0, acc0, false, false);
    }
    if constexpr ((K & 3) != 0) {             // partial tail: clamp + select
        const int ka  = KFULL + kh;
        const bool p0 = (ka     < K);
        const bool p1 = (ka + 1 < K);
        const int o0  = p0 ? (KFULL) : (-kh);         // Ap already includes +kh
        const int o1  = p1 ? (KFULL + 1) : (-kh);
        float ta0 = Ap[o0];                           // always in-bounds
        float ta1 = Ap[o1];
        v2f a0, b0;
        a0.x = p0 ? ta0 : 0.0f;
        a0.y = p1 ? ta1 : 0.0f;
        if (BT) {
            float tb0 = Bp[o0];
            float tb1 = Bp[o1];
            b0.x = p0 ? tb0 : 0.0f;
            b0.y = p1 ? tb1 : 0.0f;
        } else {
            float tb0 = Bp[(long)(p0 ? ka : 0)     * ldb];
            float tb1 = Bp[(long)(p1 ? ka + 1 : 0) * ldb];
            b0.x = p0 ? tb0 : 0.0f;
            b0.y = p1 ? tb1 : 0.0f;
        }
        acc0 = __builtin_amdgcn_wmma_f32_16x16x4_f32(
            false, a0, false, b0, (short)0, acc0, false, false);
    }

    const float bval = BIAS ? bias[ncol] : 0.0f;
    const int rbase = row0 + ((lane >> 4) << 3);
    #pragma unroll
    for (int r = 0; r < 8; ++r) {
        float v = acc0[r] + acc1[r] + bval;
        if (RELU) v = fmaxf(v, 0.0f);
        C[zC + (long)(rbase + r) * ldc + ncol] = v;
    }
}

// ---------------------------------------------------------------------------
// Reductions over j = 0..63 (two wave32's): shuffle within wave, LDS combine.
// ---------------------------------------------------------------------------
__device__ inline float rowReduceSum(float v, int il, int j, float* part) {
    #pragma unroll
    for (int m = 16; m > 0; m >>= 1) v += __shfl_xor(v, m, 32);
    const int wid = (il << 1) | (j >> 5);
    if ((j & 31) == 0) part[wid] = v;
    __syncthreads();
    float r = part[il << 1] + part[(il << 1) | 1];
    __syncthreads();
    return r;
}
__device__ inline float rowReduceMax(float v, int il, int j, float* part) {
    #pragma unroll
    for (int m = 16; m > 0; m >>= 1) v = fmaxf(v, __shfl_xor(v, m, 32));
    const int wid = (il << 1) | (j >> 5);
    if ((j & 31) == 0) part[wid] = v;
    __syncthreads();
    float r = fmaxf(part[il << 1], part[(il << 1) | 1]);
    __syncthreads();
    return r;
}

// ---------------------------------------------------------------------------
// Attention scores + masked softmax + probs / pw / pp.  One block per hb.
// ---------------------------------------------------------------------------
__global__ __launch_bounds__(256) void scores_kernel(
    const float* __restrict__ U, const float* __restrict__ V,
    const float* __restrict__ P, const float* __restrict__ path,
    const float* __restrict__ mask,
    const float* __restrict__ bh, const float* __restrict__ wo,
    const float* __restrict__ bo_p,
    float* __restrict__ probs, float* __restrict__ pw, float* __restrict__ pp)
{
    __shared__ float Ul[64 * 64];
    __shared__ float Vl[64 * 65];
    __shared__ float bhl[64], wol[64];
    __shared__ float part[8];

    const int tid = threadIdx.x;
    const int hb  = blockIdx.x;
    const int b   = hb & 31;
    const int il  = tid >> 6;
    const int j   = tid & 63;

    for (int idx = tid; idx < 4096; idx += 256) {
        const int ii = idx >> 6, kk = idx & 63;
        Ul[idx] = U[(long)hb * 4096 + idx];
        Vl[ii * 65 + kk] = V[(long)hb * 4096 + idx];
    }
    if (tid < 64) { bhl[tid] = bh[tid]; wol[tid] = wo[tid]; }
    const float bo = bo_p[0];
    __syncthreads();

    for (int i0 = 0; i0 < 64; i0 += 4) {
        const int i = i0 + il;
        const float* Prow = P + ((((long)b * 64 + i) * 64 + j) << 6);
        const float* Urow = Ul + i * 64;
        const float* Vrow = Vl + j * 65;
        float sc = 0.0f;
        #pragma unroll 8
        for (int k = 0; k < 64; ++k) {
            float s = Urow[k] + Vrow[k] + Prow[k] + bhl[k];
            s = (s > 0.0f) ? s : 0.2f * s;              // leaky_relu(0.2)
            sc = fmaf(s, wol[k], sc);
        }
        const float mk = mask[((long)b * 64 + i) * 64 + j];
        sc = (sc + bo) * mk;

        const float mx  = rowReduceMax(sc, il, j, part);
        const float e   = expf(sc - mx) * mk;
        const float sum = rowReduceSum(e, il, j, part);
        const float p   = e / (sum + 1e-20f) * mk;

        probs[(((long)hb * 64 + i) << 6) + j] = p;

        const float sp = rowReduceSum(p, il, j, part);
        if (j == 0) pw[hb * 64 + i] = sp;

        const float* prow = path + (((long)b * 64 + i) * 64 + j) * 10;
        for (int f = 0; f < 10; ++f) {
            const float s = rowReduceSum(p * prow[f], il, j, part);
            if (j == 0) pp[(hb * 64 + i) * 16 + f] = s;
        }
    }
}

// ---------------------------------------------------------------------------
// Elementwise kernels
// ---------------------------------------------------------------------------
__global__ void remap_h0(const float* __restrict__ h0raw,
                         float* __restrict__ h0, float* __restrict__ atom)
{
    const int t = blockIdx.x * 256 + threadIdx.x;     // 524288 total
    const int k = t & 63, a = (t >> 6) & 63, hb = t >> 12;
    const int h = hb >> 5, b = hb & 31;
    const float v = h0raw[(long)(b * 64 + a) * 256 + h * 64 + k];
    h0[t] = v;
    atom[t] = v;
}

__global__ void combine_kernel(
    const float* __restrict__ pwv, const float* __restrict__ ppv,
    const float* __restrict__ AM1, const float* __restrict__ MSG2,
    const float* __restrict__ Wm, const float* __restrict__ bm,
    const float* __restrict__ h0, float* __restrict__ atom_next)
{
    const int t = blockIdx.x * 256 + threadIdx.x;     // 524288 total
    const int k = t & 63;
    const int ik = t >> 6;                            // hb*64 + i
    float v = pwv[ik] * AM1[t] + MSG2[t] + bm[k] + h0[t];
    const float* ppr = ppv + (long)ik * 16;
    const float* wr  = Wm + (long)k * 138 + 128;      // Mp block
    #pragma unroll
    for (int f = 0; f < 10; ++f) v = fmaf(ppr[f], wr[f], v);
    atom_next[t] = fmaxf(v, 0.0f);                    // relu(attn_h + h0)
}

__global__ void mean_attn(const float* __restrict__ probs, float* __restrict__ outp)
{
    const int t = blockIdx.x * 256 + threadIdx.x;     // 131072 total
    const int j = t & 63, i = (t >> 6) & 63, b = t >> 12;
    float s = 0.0f;
    #pragma unroll
    for (int h = 0; h < 4; ++h)
        s += probs[(((long)((h << 5) | b) * 64 + i) << 6) + j];
    outp[t] = 0.25f * s;
}

__global__ void pack_final(const float* __restrict__ atom_input,
                           const float* __restrict__ atomh, float* __restrict__ Xf)
{
    const int t = blockIdx.x * 256 + threadIdx.x;     // 2048*338 = 692224
    const int m = t / 338, c = t % 338;
    float v;
    if (c < 82) {
        v = atom_input[(long)m * 82 + c];
    } else {
        const int cc = c - 82, h = cc >> 6, k = cc & 63;
        const int b = m >> 6, a = m & 63;
        v = atomh[((long)(h * 32 + b) * 64 + a) * 64 + k];
    }
    Xf[t] = v;
}

// ---------------------------------------------------------------------------
extern "C" void kernel_launch(void* const* d_in, const int* in_sizes, int n_in,
                              void* d_out, int out_size, void* d_ws, size_t ws_size,
                              hipStream_t stream)
{
    const float* atom_input = (const float*)d_in[0];   // [2048, 82]
    const float* path_input = (const float*)d_in[1];   // [32,64,64,10]
    const float* attn_mask  = (const float*)d_in[2];   // [32,64,64]
    const float* W_atom_i   = (const float*)d_in[3];   // [256, 82]
    const float* W_attn_h_w = (const float*)d_in[4];   // [64, 138]
    const float* W_attn_h_b = (const float*)d_in[5];   // [64]
    const float* W_attn_o_w = (const float*)d_in[6];   // [1, 64]
    const float* W_attn_o_b = (const float*)d_in[7];   // [1]
    const float* W_msg_h_w  = (const float*)d_in[8];   // [64, 138]
    const float* W_msg_h_b  = (const float*)d_in[9];   // [64]
    const float* W_atom_o_w = (const float*)d_in[10];  // [256, 338]
    const float* W_atom_o_b = (const float*)d_in[11];  // [256]
    float* out = (float*)d_out;                        // 524288 + 2*131072

    float* ws = (float*)d_ws;
    size_t off = 0;
    float* P      = ws + off; off += (size_t)131072 * 64;  // path @ Wp^T
    float* h0raw  = ws + off; off += 2048 * 256;
    float* h0     = ws + off; off += 524288;
    float* atomA  = ws + off; off += 524288;
    float* atomB  = ws + off; off += 524288;
    float* U      = ws + off; off += 524288;
    float* V      = ws + off; off += 524288;
    float* AM1    = ws + off; off += 524288;
    float* AM2    = ws + off; off += 524288;
    float* MSG2   = ws + off; off += 524288;
    float* probs0 = ws + off; off += 524288;
    float* probs1 = ws + off; off += 524288;
    float* pwv    = ws + off; off += 8192;
    float* ppv    = ws + off; off += 8192 * 16;
    float* Xf     = ws + off; off += (size_t)2048 * 338;

    // h0_raw = atom_input @ W_atom_i^T   [2048 x 256], K=82
    gemm_f32_wmma<82, true, false, false><<<dim3(2048 / 64, 256 / 16, 1), 128, 0, stream>>>(
        atom_input, 82, 0, W_atom_i, 82, 0, h0raw, 256, 0, nullptr);
    remap_h0<<<2048, 256, 0, stream>>>(h0raw, h0, atomA);

    // P = path @ Wp^T   [131072 x 64], K=10 (layer-invariant)
    gemm_f32_wmma<10, true, false, false><<<dim3(131072 / 64, 64 / 16, 1), 128, 0, stream>>>(
        path_input, 10, 0, W_attn_h_w + 128, 138, 0, P, 64, 0, nullptr);

    float* cur = atomA;
    float* nxt = atomB;
    for (int layer = 0; layer < 2; ++layer) {
        float* probsL = layer ? probs1 : probs0;
        // Four [8192 x 64] x [64 x 64] projections of atom_h
        gemm_f32_wmma<64, true, false, false><<<dim3(8192 / 64, 4, 1), 128, 0, stream>>>(
            cur, 64, 0, W_attn_h_w + 0, 138, 0, U, 64, 0, nullptr);
        gemm_f32_wmma<64, true, false, false><<<dim3(8192 / 64, 4, 1), 128, 0, stream>>>(
            cur, 64, 0, W_attn_h_w + 64, 138, 0, V, 64, 0, nullptr);
        gemm_f32_wmma<64, true, false, false><<<dim3(8192 / 64, 4, 1), 128, 0, stream>>>(
            cur, 64, 0, W_msg_h_w + 0, 138, 0, AM1, 64, 0, nullptr);
        gemm_f32_wmma<64, true, false, false><<<dim3(8192 / 64, 4, 1), 128, 0, stream>>>(
            cur, 64, 0, W_msg_h_w + 64, 138, 0, AM2, 64, 0, nullptr);

        scores_kernel<<<128, 256, 0, stream>>>(
            U, V, P, path_input, attn_mask,
            W_attn_h_b, W_attn_o_w, W_attn_o_b, probsL, pwv, ppv);

        // MSG2 = probs @ AM2   (batched per hb: 128 x [64x64x64])
        gemm_f32_wmma<64, false, false, false><<<dim3(1, 4, 128), 128, 0, stream>>>(
            probsL, 64, 4096, AM2, 64, 4096, MSG2, 64, 4096, nullptr);

        combine_kernel<<<2048, 256, 0, stream>>>(
            pwv, ppv, AM1, MSG2, W_msg_h_w, W_msg_h_b, h0, nxt);

        float* t = cur; cur = nxt; nxt = t;
    }

    // attention maps (mean over heads) -> outputs 2 and 3
    mean_attn<<<512, 256, 0, stream>>>(probs0, out + 524288);
    mean_attn<<<512, 256, 0, stream>>>(probs1, out + 524288 + 131072);

    // out = relu([atom_input | final] @ W_atom_o^T + b)
    pack_final<<<2704, 256, 0, stream>>>(atom_input, cur, Xf);
    gemm_f32_wmma<338, true, true, true><<<dim3(2048 / 64, 256 / 16, 1), 128, 0, stream>>>(
        Xf, 338, 0, W_atom_o_w, 338, 0, out, 256, 0, W_atom_o_b);

    (void)in_sizes; (void)n_in; (void)out_size; (void)ws_size;
}